// PointCloud_CSDF_84224308674625
// MI455X (gfx1250) — compile-verified
//
#include <hip/hip_runtime.h>
#include <math.h>

// CDNA5 (gfx1250) — wave32, WMMA f32 16x16x4.
typedef __attribute__((ext_vector_type(2))) float v2f;
typedef __attribute__((ext_vector_type(8))) float v8f;

#define NSPLIT 8   // split the N (point) dimension across waves; flat min makes this free
#define WS_B_OFF 256  // byte offset of packed-B region in workspace (keeps 8 min slots + alignment)

// acc = min(acc, x, y) in one VOP3, no fminf canonicalize overhead
#define MIN3(acc, x, y) \
    asm("v_min3_num_f32 %0, %1, %2, %0" : "+v"(acc) : "v"(x), "v"(y))

// --- ordered-uint encoding so unsigned atomicMin == float min (handles negatives) ---
__device__ __forceinline__ unsigned float_to_ordered(float f) {
    unsigned u = __float_as_uint(f);
    return u ^ ((unsigned)((int)u >> 31) | 0x80000000u);
}
__device__ __forceinline__ float ordered_to_float(unsigned k) {
    unsigned u = (k & 0x80000000u) ? (k ^ 0x80000000u) : ~k;
    return __uint_as_float(u);
}

__global__ void csdf_init(unsigned* __restrict__ ws, int nb) {
    int i = threadIdx.x;
    if (i < nb) ws[i] = 0xFFFFFFFFu;  // encodes +max
}

// Pack the WMMA B-operand lane layout once per launch:
//   B col n = [px, py, pz, p2];  lanes 0-15 hold (K0,K1)=(px,py), lanes 16-31 hold (K2,K3)=(pz,p2)
__global__ void csdf_packB(const float* __restrict__ pcd, v2f* __restrict__ wsB, int ntiles) {
    const int idx  = blockIdx.x * blockDim.x + threadIdx.x;  // (tile, lane) flattened
    const int lane = idx & 31;
    const int t    = idx >> 5;
    if (t >= ntiles) return;
    const int   n  = t * 16 + (lane & 15);
    const float px = pcd[n * 3 + 0], py = pcd[n * 3 + 1], pz = pcd[n * 3 + 2];
    const float p2 = __builtin_fmaf(px, px, __builtin_fmaf(py, py, pz * pz));
    v2f b;
    b.x = (lane < 16) ? px : pz;
    b.y = (lane < 16) ? py : p2;
    wsB[idx] = b;
}

__device__ __forceinline__ v2f make_b_inline(const float* __restrict__ p3, int lane) {
    const float px = p3[0], py = p3[1], pz = p3[2];
    const float p2 = __builtin_fmaf(px, px, __builtin_fmaf(py, py, pz * pz));
    v2f b;
    b.x = (lane < 16) ? px : pz;
    b.y = (lane < 16) ? py : p2;
    return b;
}

// Each wave owns TWO 16-query tiles and one N-chunk.
//   A row m = [-2qx, -2qy, -2qz, 1],  B col n = [px, py, pz, p2],  C[m][n] = q2[m]
//   => D[m][n] = q2 + p2 - 2 q.p  (exact fp32, matches reference numerics)
// csdf[b] = sqrt(max(min_{q,n} d2, 0)) - radius, so only a flat per-batch min survives.
template <bool PACKED>
__global__ __launch_bounds__(256) void csdf_main(const float* __restrict__ x,
                                                 const float* __restrict__ pcd,
                                                 const v2f* __restrict__ wsB,
                                                 unsigned* __restrict__ ws,
                                                 int M, int P, int N, int iters) {
    const int lane  = threadIdx.x & 31;
    const int gwave = (blockIdx.x * blockDim.x + threadIdx.x) >> 5;
    const int group = gwave / NSPLIT;          // 32-query group (two 16-row tiles)
    const int chunk = gwave - group * NSPLIT;
    const int mbase = group * 32;
    if (mbase >= M) return;                    // wave-uniform: EXEC stays all-1s for live waves
    const int batch = mbase / P;               // groups never straddle batches (P % 32 == 0)

    // ---- A matrices (once per wave): lanes 0-15 hold K0/K1, lanes 16-31 hold K2/K3
    const int   m0 = mbase + (lane & 15);
    const int   m1 = m0 + 16;
    const float q0x = x[m0 * 3 + 0], q0y = x[m0 * 3 + 1], q0z = x[m0 * 3 + 2];
    const float q1x = x[m1 * 3 + 0], q1y = x[m1 * 3 + 1], q1z = x[m1 * 3 + 2];
    v2f a0, a1;
    a0.x = (lane < 16) ? (-2.0f * q0x) : (-2.0f * q0z);
    a0.y = (lane < 16) ? (-2.0f * q0y) : 1.0f;
    a1.x = (lane < 16) ? (-2.0f * q1x) : (-2.0f * q1z);
    a1.y = (lane < 16) ? (-2.0f * q1y) : 1.0f;

    // ---- C matrices: c[r] = q2 of tile row (r + 8*(lane>=16))
    const float q20 = q0x * q0x + q0y * q0y + q0z * q0z;
    const float q21 = q1x * q1x + q1y * q1y + q1z * q1z;
    v8f c0, c1;
#pragma unroll
    for (int r = 0; r < 8; ++r) {
        const int src = r + ((lane & 16) >> 1);
        c0[r] = __shfl(q20, src, 32);
        c1[r] = __shfl(q21, src, 32);
    }

    // ---- 8 running-min accumulators (4 per tile via min3 folding)
    float acc[8];
#pragma unroll
    for (int r = 0; r < 8; ++r) acc[r] = 3.402823466e38f;

    const int nbeg = chunk * (iters * 16);

    // one compute step: B chunk -> 2 WMMAs -> 8 min3
    auto step = [&](v2f b) {
        v8f d0 = __builtin_amdgcn_wmma_f32_16x16x4_f32(false, a0, false, b,
                                                       (short)0, c0, false, false);
        v8f d1 = __builtin_amdgcn_wmma_f32_16x16x4_f32(false, a1, false, b,
                                                       (short)0, c1, false, false);
        MIN3(acc[0], d0[0], d0[1]);
        MIN3(acc[1], d0[2], d0[3]);
        MIN3(acc[2], d0[4], d0[5]);
        MIN3(acc[3], d0[6], d0[7]);
        MIN3(acc[4], d1[0], d1[1]);
        MIN3(acc[5], d1[2], d1[3]);
        MIN3(acc[6], d1[4], d1[5]);
        MIN3(acc[7], d1[6], d1[7]);
    };

    if (PACKED) {
        const v2f* __restrict__ bp = wsB + (size_t)(nbeg / 16) * 32 + lane;
        v2f b = *bp;                               // one-deep prefetch pipeline
        for (int i = 1; i < iters; ++i) {
            bp += 32;
            v2f bn = *bp;
            step(b);
            b = bn;
        }
        step(b);
    } else {
        const float* __restrict__ pp = pcd + (size_t)(nbeg + (lane & 15)) * 3;
        v2f b = make_b_inline(pp, lane);
        for (int i = 1; i < iters; ++i) {
            pp += 48;                              // 16 points * 3 floats
            v2f bn = make_b_inline(pp, lane);
            step(b);
            b = bn;
        }
        step(b);
    }

    // ---- wave-local flat min: 8 accs -> 1 scalar -> lanes
    float m = fminf(acc[0], acc[1]);
    m = fminf(m, fminf(acc[2], acc[3]));
    m = fminf(m, fminf(acc[4], acc[5]));
    m = fminf(m, fminf(acc[6], acc[7]));
#pragma unroll
    for (int off = 16; off >= 1; off >>= 1)
        m = fminf(m, __shfl_xor(m, off, 32));

    if (lane == 0)
        atomicMin(&ws[batch], float_to_ordered(m));
}

__global__ void csdf_final(const unsigned* __restrict__ ws,
                           const float* __restrict__ radius,
                           float* __restrict__ out, int nb) {
    int i = threadIdx.x;
    if (i < nb) {
        float d2 = fmaxf(ordered_to_float(ws[i]), 0.0f);
        out[i] = sqrtf(d2) - radius[0];
    }
}

extern "C" void kernel_launch(void* const* d_in, const int* in_sizes, int n_in,
                              void* d_out, int out_size, void* d_ws, size_t ws_size,
                              hipStream_t stream) {
    const float* x      = (const float*)d_in[0];
    const float* pcd    = (const float*)d_in[1];
    const float* radius = (const float*)d_in[2];
    float*       out    = (float*)d_out;
    unsigned*    wsMin  = (unsigned*)d_ws;
    v2f*         wsB    = (v2f*)((char*)d_ws + WS_B_OFF);

    const int M = in_sizes[0] / 3;   // 32768 queries
    const int N = in_sizes[1] / 3;   // 8192 points
    const int B = out_size;          // 8 batches
    const int P = M / B;             // 4096 queries per batch

    const int ntiles     = N / 16;                    // B tiles along the point dim
    const int iters      = N / (16 * NSPLIT);         // chunks per wave
    const int groups     = (M + 31) / 32;             // 32-query groups
    const int totalWaves = groups * NSPLIT;
    const int threads    = 256;                       // 8 wave32s per block
    const int blocks     = (totalWaves * 32 + threads - 1) / threads;

    const size_t needB = (size_t)WS_B_OFF + (size_t)ntiles * 32 * sizeof(v2f);
    const bool packed  = ws_size >= needB;            // constant across calls -> deterministic

    csdf_init<<<1, 32, 0, stream>>>(wsMin, B);
    if (packed) {
        const int pthreads = 256;
        const int pblocks  = (ntiles * 32 + pthreads - 1) / pthreads;
        csdf_packB<<<pblocks, pthreads, 0, stream>>>(pcd, wsB, ntiles);
        csdf_main<true><<<blocks, threads, 0, stream>>>(x, pcd, wsB, wsMin, M, P, N, iters);
    } else {
        csdf_main<false><<<blocks, threads, 0, stream>>>(x, pcd, wsB, wsMin, M, P, N, iters);
    }
    csdf_final<<<1, 32, 0, stream>>>(wsMin, radius, out, B);
}